// slicedKLclass_78460462563934
// MI455X (gfx1250) — compile-verified
//
#include <hip/hip_runtime.h>
#include <hip/hip_bf16.h>

// -------- CDNA5 WMMA types --------
typedef __attribute__((ext_vector_type(2))) float v2f;
typedef __attribute__((ext_vector_type(8))) float v8f;

// Problem constants (from reference)
#define NN     8192
#define DD     512
#define SS     64
#define KK     8
#define HH     240
#define SK     512   // S*K, projection output channels

// ---------------- CDNA5 async global->LDS helpers (inline asm) ----------------
// global_load_async_to_lds_b128 vdst(LDS offset VGPR), vaddr(64-bit), off
// Tracked by ASYNCcnt; loads complete in order, so waiting ASYNCcnt<=N releases
// the oldest copies first (used for the double-buffer pipeline below).
__device__ __forceinline__ void async_copy_b128(void* lds_dst, const float* gsrc) {
  uint32_t lds_off = (uint32_t)(size_t)lds_dst;   // low 32 bits = LDS offset
  asm volatile("global_load_async_to_lds_b128 %0, %1, off"
               :: "v"(lds_off), "v"(gsrc) : "memory");
}
__device__ __forceinline__ void wait_async_le6() {
  asm volatile("s_wait_asynccnt 0x6" ::: "memory");
}
__device__ __forceinline__ void wait_async_0() {
  asm volatile("s_wait_asynccnt 0x0" ::: "memory");
}

// ---------------- Phase 1: projection GEMM ----------------
// sample = In[8192,512] * thetas^T   (thetas viewed as [512 c][512 d], row-major)
// Workgroup: 256 threads = 8 waves. Tile: 128(M) x 64(N), K-chunk 32.
// Double-buffered LDS tiles, filled with async b128 copies (row-major,
// row stride 36 floats = 144 B: 16B-aligned rows for b128, and conflict-free
// b64 fragment reads: banks {4j,4j+1} for lanes 0-15, {4j+2,4j+3} for 16-31).
#define PM 128
#define PN 64
#define PK 32
#define LSTR 36   // LDS row stride in floats

__global__ __launch_bounds__(256)
void proj_kernel(const float* __restrict__ X, const float* __restrict__ Y,
                 const float* __restrict__ T,
                 float* __restrict__ XS, float* __restrict__ YS) {
  __shared__ float lA[2][PM * LSTR];   // 2 x 18 KB
  __shared__ float lB[2][PN * LSTR];   // 2 x  9 KB

  const float* In  = (blockIdx.z == 0) ? X : Y;
  float*       Out = (blockIdx.z == 0) ? XS : YS;

  const int m0   = blockIdx.x * PM;
  const int c0   = blockIdx.y * PN;
  const int tid  = threadIdx.x;
  const int lane = tid & 31;
  const int wave = tid >> 5;
  const int half = lane >> 4;    // 0: lanes 0-15 (K0,K1), 1: lanes 16-31 (K2,K3)
  const int l15  = lane & 15;
  const int wm   = wave * 16;    // wave's M sub-tile within the 128-row block

  // per-thread async-copy assignments (uniform trip counts, 6 instr/wave/chunk)
  // A tile: 128 rows x 8 b128 chunks = 1024 ops -> 4 per thread
  // B tile:  64 rows x 8 b128 chunks =  512 ops -> 2 per thread
  auto issue_chunk = [&](int buf, int kb) {
    #pragma unroll
    for (int i = 0; i < 4; ++i) {
      int idx = tid + i * 256;
      int m   = idx >> 3;
      int q   = idx & 7;                     // 4-float chunk within row
      async_copy_b128(&lA[buf][m * LSTR + q * 4],
                      In + (size_t)(m0 + m) * DD + kb + q * 4);
    }
    #pragma unroll
    for (int i = 0; i < 2; ++i) {
      int idx = tid + i * 256;
      int c   = idx >> 3;
      int q   = idx & 7;
      async_copy_b128(&lB[buf][c * LSTR + q * 4],
                      T + (size_t)(c0 + c) * DD + kb + q * 4);
    }
  };

  v8f acc[4] = {};               // 16(M) x 64(N) per wave

  const int nchunks = DD / PK;   // 16
  issue_chunk(0, 0);

  for (int ci = 0; ci < nchunks; ++ci) {
    const int cur = ci & 1;
    const bool more = (ci + 1) < nchunks;
    if (more) issue_chunk(cur ^ 1, (ci + 1) * PK);   // prefetch next chunk

    // release the 6 copies belonging to `cur` (in-order completion)
    if (more) wait_async_le6(); else wait_async_0();
    __syncthreads();

    // ---- 8 WMMA K-steps (K=4 each) on the current buffer ----
    #pragma unroll
    for (int k0 = 0; k0 < PK; k0 += 4) {
      v2f a = *(const v2f*)&lA[cur][(wm + l15) * LSTR + k0 + 2 * half];
      #pragma unroll
      for (int j = 0; j < 4; ++j) {
        v2f b = *(const v2f*)&lB[cur][(j * 16 + l15) * LSTR + k0 + 2 * half];
        acc[j] = __builtin_amdgcn_wmma_f32_16x16x4_f32(
            false, a, false, b, (short)0, acc[j], false, false);
      }
    }
    __syncthreads();   // everyone done reading `cur` before it is refilled
  }

  // ---- epilogue: C/D layout: VGPR r -> row r (lanes 0-15) / r+8 (16-31) ----
  #pragma unroll
  for (int j = 0; j < 4; ++j) {
    #pragma unroll
    for (int r = 0; r < 8; ++r) {
      int row = m0 + wm + half * 8 + r;
      int col = c0 + j * 16 + l15;
      Out[(size_t)row * SK + col] = acc[j][r];
    }
  }
}

// ---------------- Phase 2: per-slice MLP via WMMA ----------------
// One wave handles (slice s, 16 rows of n) for BOTH x and y (reuses W1 loads).
// pred[n] = sum_h relu(sum_k sample[n, s*8+k]*W1[s,h,k] + b1[s,h]) * W3[s,h] + b3[s]
__global__ __launch_bounds__(256)
void mlp_kernel(const float* __restrict__ XS, const float* __restrict__ YS,
                const float* __restrict__ W1, const float* __restrict__ B1,
                const float* __restrict__ W3, const float* __restrict__ B3,
                float* __restrict__ PX, float* __restrict__ PY) {
  const int tid  = threadIdx.x;
  const int lane = tid & 31;
  const int wave = tid >> 5;
  const int gw   = blockIdx.x * 8 + wave;   // global wave id
  const int s    = gw >> 9;                 // / (8192/16)
  const int nb   = gw & 511;
  const int n0   = nb * 16;
  const int half = lane >> 4;
  const int l15  = lane & 15;

  const float* w1s = W1 + (size_t)s * HH * KK;
  const float* b1s = B1 + (size_t)s * HH;
  const float* w3s = W3 + (size_t)s * HH;
  const size_t aoff = (size_t)(n0 + l15) * SK + s * KK;  // A row base for this lane

  v8f accx = {};
  v8f accy = {};

  for (int ht = 0; ht < HH / 16; ++ht) {    // 15 H-tiles of 16
    const int h    = ht * 16 + l15;
    const float bias = b1s[h];
    const float w3v  = w3s[h];
    v8f hx = {};
    v8f hy = {};
    #pragma unroll
    for (int kk = 0; kk < KK; kk += 4) {    // K=8 -> 2 WMMA steps
      int kp = kk + 2 * half;               // this lane's (K,K+1) pair start
      v2f ax = *(const v2f*)(XS + aoff + kp);
      v2f ay = *(const v2f*)(YS + aoff + kp);
      v2f b  = *(const v2f*)(w1s + (size_t)h * KK + kp);  // W1[s,h,kp..kp+1]
      hx = __builtin_amdgcn_wmma_f32_16x16x4_f32(false, ax, false, b, (short)0, hx, false, false);
      hy = __builtin_amdgcn_wmma_f32_16x16x4_f32(false, ay, false, b, (short)0, hy, false, false);
    }
    #pragma unroll
    for (int r = 0; r < 8; ++r) {
      float vx = hx[r] + bias;
      float vy = hy[r] + bias;
      accx[r] += (vx > 0.0f ? vx : 0.0f) * w3v;
      accy[r] += (vy > 0.0f ? vy : 0.0f) * w3v;
    }
  }

  // reduce over the 16 H-columns held by the 16 lanes of each half-wave
  #pragma unroll
  for (int r = 0; r < 8; ++r) {
    float vx = accx[r];
    float vy = accy[r];
    #pragma unroll
    for (int m = 1; m < 16; m <<= 1) {
      vx += __shfl_xor(vx, m, 32);
      vy += __shfl_xor(vy, m, 32);
    }
    accx[r] = vx;
    accy[r] = vy;
  }

  if (l15 == 0) {
    const float bb = B3[s];
    #pragma unroll
    for (int r = 0; r < 8; ++r) {
      int n = n0 + half * 8 + r;
      PX[(size_t)s * NN + n] = accx[r] + bb;
      PY[(size_t)s * NN + n] = accy[r] + bb;
    }
  }
}

// ---------------- Phase 3: per-slice reduction ----------------
// ret[s] = mean_n(pred_x) - (logsumexp_n(pred_y) - log(N))
__global__ __launch_bounds__(256)
void reduce_kernel(const float* __restrict__ PX, const float* __restrict__ PY,
                   float* __restrict__ RET) {
  const int s   = blockIdx.x;
  const int tid = threadIdx.x;
  const float* px = PX + (size_t)s * NN;
  const float* py = PY + (size_t)s * NN;

  float sum = 0.0f;
  float m = -__builtin_inff();
  float l = 0.0f;
  for (int n = tid; n < NN; n += 256) {
    sum += px[n];
    float v  = py[n];
    float nm = fmaxf(m, v);
    l = l * expf(m - nm) + expf(v - nm);
    m = nm;
  }

  __shared__ float sS[256], sM[256], sL[256];
  sS[tid] = sum; sM[tid] = m; sL[tid] = l;
  __syncthreads();
  for (int st = 128; st > 0; st >>= 1) {
    if (tid < st) {
      sS[tid] += sS[tid + st];
      float m1 = sM[tid], m2 = sM[tid + st];
      float nm = fmaxf(m1, m2);
      sL[tid] = sL[tid] * expf(m1 - nm) + sL[tid + st] * expf(m2 - nm);
      sM[tid] = nm;
    }
    __syncthreads();
  }
  if (tid == 0) {
    float meanx = sS[0] / (float)NN;
    float lse   = logf(sL[0]) + sM[0] - logf((float)NN);
    RET[s] = meanx - lse;
  }
}

// ---------------- Phase 4: final scalar ----------------
__global__ void finalize_kernel(const float* __restrict__ RET, float* __restrict__ out) {
  __shared__ float sv[64];
  const int t = threadIdx.x;   // 64 threads
  sv[t] = -RET[t];
  __syncthreads();
  for (int st = 32; st > 0; st >>= 1) {
    if (t < st) sv[t] += sv[t + st];
    __syncthreads();
  }
  if (t == 0) out[0] = sv[0] / (float)SS;
}

// ---------------- launch ----------------
extern "C" void kernel_launch(void* const* d_in, const int* in_sizes, int n_in,
                              void* d_out, int out_size, void* d_ws, size_t ws_size,
                              hipStream_t stream) {
  const float* X      = (const float*)d_in[0];   // [8192,512]
  const float* Y      = (const float*)d_in[1];   // [8192,512]
  // d_in[2] = X_noise, unused (sigma_noise == 0)
  const float* thetas = (const float*)d_in[3];   // [64,8,512] -> [512,512]
  const float* W1     = (const float*)d_in[4];   // [64,240,8]
  const float* B1     = (const float*)d_in[5];   // [64,240]
  const float* W3     = (const float*)d_in[6];   // [64,240]
  const float* B3     = (const float*)d_in[7];   // [64]

  // workspace layout (floats): x_sample, y_sample, pred_x, pred_y, ret  (~38 MB)
  float* ws = (float*)d_ws;
  float* xs  = ws;
  float* ys  = xs + (size_t)NN * SK;
  float* px  = ys + (size_t)NN * SK;
  float* py  = px + (size_t)SS * NN;
  float* ret = py + (size_t)SS * NN;

  dim3 g1(NN / PM, SK / PN, 2);                       // (64, 8, 2)
  proj_kernel<<<g1, 256, 0, stream>>>(X, Y, thetas, xs, ys);

  dim3 g2((SS * (NN / 16)) / 8);                      // 4096 blocks of 8 waves
  mlp_kernel<<<g2, 256, 0, stream>>>(xs, ys, W1, B1, W3, B3, px, py);

  reduce_kernel<<<SS, 256, 0, stream>>>(px, py, ret);
  finalize_kernel<<<1, 64, 0, stream>>>(ret, (float*)d_out);
}